// GraphGRUEncoder_7043746365717
// MI455X (gfx1250) — compile-verified
//
#include <hip/hip_runtime.h>

// GraphGRUEncoder for MI455X (gfx1250, wave32).
// N=50000 nodes, T=12 steps, F=2 in-feats, H=64 hidden, E=800000 edges, L=2 layers.
//
// Strategy:
//  * Exploit linearity of the GCN aggregation: agg([x,h]) = [agg(x), agg(h)].
//    Per step we scatter x_t (F cols), h (64 cols), r*h (64 cols); agg(x_t) is
//    reused by BOTH the gate GCN and the candidate GCN.
//  * fp32 WMMA (v_wmma_f32_16x16x4_f32) for the GEMMs -> matches fp32 reference
//    precision; GEMM FLOPs (~45 GFLOP total) are trivial vs scatter bandwidth.
//  * All per-step buffers (~80MB) are L2-resident (192MB L2) so edge atomics
//    and gathers hit L2, not HBM (23.3 TB/s would already make this ~ms-scale).
//  * Candidate GEMM epilogue fuses tanh + GRU update + output store.

typedef __attribute__((ext_vector_type(2))) float v2f;
typedef __attribute__((ext_vector_type(8))) float v8f;

#define HH 64

__global__ __launch_bounds__(256) void zero_kernel(float* __restrict__ p, int n) {
  int i = blockIdx.x * blockDim.x + threadIdx.x;
  if (i < n) p[i] = 0.0f;
}

// Scatter pass 1: aggX[dst] += w * x_t[src]  (cols 0..F-1)
//                 aggH[dst] += w * h[src]    (cols 0..63)
__global__ __launch_bounds__(256) void scatter_xh(
    const int* __restrict__ src, const int* __restrict__ dst,
    const float* __restrict__ ew, int E,
    const float* __restrict__ xbase, int xstride, int F,
    const float* __restrict__ hbase, int hstride,
    float* __restrict__ aggX, float* __restrict__ aggH) {
  const int C = F + HH;
  long long idx = (long long)blockIdx.x * blockDim.x + threadIdx.x;
  long long total = (long long)E * C;
  if (idx >= total) return;
  int e = (int)(idx / C);
  int c = (int)(idx % C);
  int s = src[e], d = dst[e];
  float w = ew[e];
  if (c < F) {
    atomicAdd(&aggX[(long long)d * F + c], w * xbase[(long long)s * xstride + c]);
  } else {
    int c2 = c - F;
    atomicAdd(&aggH[(long long)d * HH + c2], w * hbase[(long long)s * hstride + c2]);
  }
}

// Scatter pass 2: aggRH[dst] += w * r[src] * h[src]   (r = zr[:, 64:128])
__global__ __launch_bounds__(256) void scatter_rh(
    const int* __restrict__ src, const int* __restrict__ dst,
    const float* __restrict__ ew, int E,
    const float* __restrict__ zr,
    const float* __restrict__ hbase, int hstride,
    float* __restrict__ aggRH) {
  long long idx = (long long)blockIdx.x * blockDim.x + threadIdx.x;
  long long total = (long long)E * HH;
  if (idx >= total) return;
  int e = (int)(idx >> 6);
  int c = (int)(idx & 63);
  int s = src[e], d = dst[e];
  float r = zr[(long long)s * 128 + HH + c];
  atomicAdd(&aggRH[(long long)d * HH + c],
            ew[e] * r * hbase[(long long)s * hstride + c]);
}

// Fused GEMM:  out = act( [aggX | aggK2] @ W + bias )
//   mode 0: act = sigmoid, store to zrbuf (N x 128)           [gate GCN linear]
//   mode 1: act = tanh, then GRU update h = z*h + (1-z)*cand,
//           store to outbase (layer output slice), optionally hslot (final h)
// One wave per 16x16 output tile; A tile staged in LDS (odd stride -> no bank
// conflicts for the strided fragment reads). K handled in steps of 4 with
// v_wmma_f32_16x16x4_f32; K%4 leftover (layer0: 2) via scalar FMA.
__global__ __launch_bounds__(256) void gemm_wmma(
    const float* __restrict__ aggX, int F,
    const float* __restrict__ aggK2,            // N x 64 (aggH or aggRH)
    const float* __restrict__ W,                // (F+64) x ncols, row-major
    const float* __restrict__ bias, int ncols, int mode,
    float* __restrict__ zrbuf,
    const float* __restrict__ hbase, int hstride,
    float* __restrict__ outbase, int ostride,
    float* __restrict__ hslot) {
  __shared__ float smem[16 * 129];              // 16 x (Ktot+1), Ktot<=128
  const int Ktot = F + HH;
  const int Kp = Ktot + 1;                      // odd -> conflict-free
  const int row0 = blockIdx.x * 16;             // N = 50000 = 3125*16 exact

  // cooperative load of 16 x Ktot A-tile = [aggX | aggK2]
  for (int i = threadIdx.x; i < 16 * Ktot; i += blockDim.x) {
    int r = i / Ktot, k = i % Ktot;
    long long node = row0 + r;
    float v = (k < F) ? aggX[node * F + k] : aggK2[node * HH + (k - F)];
    smem[r * Kp + k] = v;
  }
  __syncthreads();

  const int lane = threadIdx.x & 31;
  const int wave = threadIdx.x >> 5;
  const int col0 = wave * 16;
  const int half = lane >> 4;                   // 0: lanes 0-15, 1: lanes 16-31
  const int n = lane & 15;                      // A row (M) and B col (N)
  const int kb = 2 * half;                      // K sub-offset per ISA layout

  v8f acc = {};
  const int kmax = Ktot & ~3;
  for (int k = 0; k < kmax; k += 4) {
    v2f a, b;
    // A 16x4 f32 fragment: VGPR0=K(k+kb), VGPR1=K(k+kb+1)
    a.x = smem[n * Kp + k + kb];
    a.y = smem[n * Kp + k + kb + 1];
    // B 4x16 f32 fragment: rows striped across lanes/VGPRs symmetrically
    const float* wp = W + (long long)(k + kb) * ncols + col0 + n;
    b.x = wp[0];
    b.y = wp[ncols];
    acc = __builtin_amdgcn_wmma_f32_16x16x4_f32(
        /*neg_a=*/false, a, /*neg_b=*/false, b,
        /*c_mod=*/(short)0, acc, /*reuse_a=*/false, /*reuse_b=*/false);
  }
  // leftover K (layer0: K=66 -> kk=64,65) via scalar FMA on owned C elements
  for (int kk = kmax; kk < Ktot; ++kk) {
    float wv = W[(long long)kk * ncols + col0 + n];
#pragma unroll
    for (int i = 0; i < 8; ++i) {
      int m = i + 8 * half;
      acc[i] += smem[m * Kp + kk] * wv;
    }
  }

  // epilogue (C layout: VGPR i -> M = i + 8*half, N = lane&15)
#pragma unroll
  for (int i = 0; i < 8; ++i) {
    int m = i + 8 * half;
    long long node = row0 + m;
    int col = col0 + n;
    float v = acc[i] + bias[col];
    if (mode == 0) {
      zrbuf[node * 128 + col] = 1.0f / (1.0f + __expf(-v));
    } else {
      float cand = tanhf(v);
      float z = zrbuf[node * 128 + col];           // z = zr[:, :64]
      float hp = hbase[node * (long long)hstride + col];
      float hn = z * hp + (1.0f - z) * cand;
      outbase[node * (long long)ostride + col] = hn;
      if (hslot) hslot[node * HH + col] = hn;
    }
  }
}

extern "C" void kernel_launch(void* const* d_in, const int* in_sizes, int n_in,
                              void* d_out, int out_size, void* d_ws, size_t ws_size,
                              hipStream_t stream) {
  const int N = 50000, T = 12, F0 = 2, H = 64, E = 800000;

  const float* x  = (const float*)d_in[0];
  const float* h0 = (const float*)d_in[1];            // (2, N, H) zeros-init
  const int*   ei = (const int*)d_in[2];              // (2, E)
  const float* ew = (const float*)d_in[3];
  const float* Wg[2] = {(const float*)d_in[4], (const float*)d_in[8]};
  const float* bg[2] = {(const float*)d_in[5], (const float*)d_in[9]};
  const float* Wc[2] = {(const float*)d_in[6], (const float*)d_in[10]};
  const float* bc[2] = {(const float*)d_in[7], (const float*)d_in[11]};
  const int* srcp = ei;
  const int* dstp = ei + E;

  float* out  = (float*)d_out;
  float* h1   = out;                                  // (N, H)
  float* h2   = out + (size_t)N * H;                  // (N, H)
  float* out2 = out + (size_t)2 * N * H;              // (N, T, H)

  // workspace: out1 (N*T*H) + aggX/aggH/aggRH/zr  (~218 MB total)
  float* out1    = (float*)d_ws;                      // (N, T, H)
  float* scratch = out1 + (size_t)N * T * H;

  for (int layer = 0; layer < 2; ++layer) {
    const int F = layer ? H : F0;
    const float* xptr = layer ? out1 : x;
    const int xstride = layer ? T * H : T * F0;       // row stride of x-source
    float* outBuf = layer ? out2 : out1;              // (N, T, H), stride T*H
    float* hslotF = layer ? h2 : h1;

    float* aggX  = scratch;                           // N x F
    float* aggH  = aggX  + (size_t)N * F;             // N x 64
    float* aggRH = aggH  + (size_t)N * H;             // N x 64
    float* zr    = aggRH + (size_t)N * H;             // N x 128

    for (int t = 0; t < T; ++t) {
      // hidden state lives in the previous output slice (or h0 at t==0)
      const float* hbase = (t == 0) ? (h0 + (size_t)layer * N * H)
                                    : (outBuf + (size_t)(t - 1) * H);
      const int hstride = (t == 0) ? H : T * H;
      const float* xb = xptr + (size_t)t * (layer ? H : F0);

      // 1) zero accumulators (aggX, aggH, aggRH are contiguous)
      int zn = N * (F + 2 * H);
      zero_kernel<<<(zn + 255) / 256, 256, 0, stream>>>(aggX, zn);

      // 2) scatter x_t and h over edges
      long long tot1 = (long long)E * (F + H);
      scatter_xh<<<(int)((tot1 + 255) / 256), 256, 0, stream>>>(
          srcp, dstp, ew, E, xb, xstride, F, hbase, hstride, aggX, aggH);

      // 3) gate GEMM: zr = sigmoid([aggX|aggH] @ Wg + bg), ncols=128, 8 waves
      gemm_wmma<<<N / 16, 256, 0, stream>>>(
          aggX, F, aggH, Wg[layer], bg[layer], 2 * H, /*mode=*/0,
          zr, nullptr, 0, nullptr, 0, nullptr);

      // 4) scatter r*h over edges
      long long tot2 = (long long)E * H;
      scatter_rh<<<(int)((tot2 + 255) / 256), 256, 0, stream>>>(
          srcp, dstp, ew, E, zr, hbase, hstride, aggRH);

      // 5) candidate GEMM + fused GRU update: ncols=64, 4 waves
      gemm_wmma<<<N / 16, 128, 0, stream>>>(
          aggX, F, aggRH, Wc[layer], bc[layer], H, /*mode=*/1,
          zr, hbase, hstride, outBuf + (size_t)t * H, T * H,
          (t == T - 1) ? hslotF : nullptr);
    }
  }
}